// SSIM_42090679501367
// MI455X (gfx1250) — compile-verified
//
#include <hip/hip_runtime.h>

// ---------------- problem constants ----------------
#define IMG_W 640
#define IMG_H 480
#define IMG_B 64
#define TW    64            // tile width  (output)
#define THT   32            // tile height (output)
#define LCOLS (TW + 2)      // 66: tile + halo
#define LROWS (THT + 2)     // 34
#define LSTR  68            // padded LDS row stride (floats)
#define NBX   (IMG_W / TW)  // 10
#define NBY   (IMG_H / THT) // 15
#define NPART (NBX * NBY * IMG_B) // 9600 partial sums

// Gaussian taps: g = [GA, GB, GA]; w = outer(g,g)  (separable, symmetric)
#define GA 0.0113f
#define GB 0.0838f
#define C1 0.0001f   // 0.01^2
#define C2 0.0009f   // 0.03^2

typedef __attribute__((ext_vector_type(2))) float v2f;
typedef __attribute__((ext_vector_type(8))) float v8f;

// ---- CDNA5 async global->LDS copy (ASYNCcnt path), GVS addressing ----
__device__ __forceinline__ void async_ld_f32(const float* gbase, unsigned byte_off,
                                             unsigned lds_byte) {
  asm volatile("global_load_async_to_lds_b32 %0, %1, %2"
               :: "v"(lds_byte), "v"(byte_off), "s"((unsigned long long)gbase)
               : "memory");
}
__device__ __forceinline__ void wait_async0() {
  asm volatile("s_wait_asynccnt 0" ::: "memory");
}

// Horizontal (row) 3-tap filter of x, y, x*x, y*y, x*y at one LDS row.
__device__ __forceinline__ void hrow(const float* sx, const float* sy, int lr, int c,
                                     float& ox, float& oy, float& oxx, float& oyy,
                                     float& oxy) {
  const float* px = sx + lr * LSTR + c;
  const float* py = sy + lr * LSTR + c;
  float x0 = px[0], x1 = px[1], x2 = px[2];
  float y0 = py[0], y1 = py[1], y2 = py[2];
  ox  = GA * (x0 + x2) + GB * x1;
  oy  = GA * (y0 + y2) + GB * y1;
  oxx = GA * (x0 * x0 + x2 * x2) + GB * (x1 * x1);
  oyy = GA * (y0 * y0 + y2 * y2) + GB * (y1 * y1);
  oxy = GA * (x0 * y0 + x2 * y2) + GB * (x1 * y1);
}

__global__ __launch_bounds__(256)
void ssim_tile_kernel(const float* __restrict__ x, const float* __restrict__ y,
                      float* __restrict__ partial) {
  __shared__ float sx[LROWS * LSTR];
  __shared__ float sy[LROWS * LSTR];
  __shared__ float red[8];

  const int tid = threadIdx.x;
  const int gx0 = blockIdx.x * TW;
  const int gy0 = blockIdx.y * THT;
  const size_t img_off = (size_t)blockIdx.z * (size_t)(IMG_W * IMG_H);
  const float* xb = x + img_off;
  const float* yb = y + img_off;

  const unsigned sx_base = (unsigned)(unsigned long long)&sx[0];
  const unsigned sy_base = (unsigned)(unsigned long long)&sy[0];

  // Stage tile + halo into LDS via async global->LDS (only in-range lanes issue).
  for (int i = tid; i < LROWS * LCOLS; i += 256) {
    int row = i / LCOLS;
    int col = i - row * LCOLS;
    int gr = gy0 - 1 + row;
    int gc = gx0 - 1 + col;
    unsigned e = (unsigned)(row * LSTR + col) * 4u;
    if ((unsigned)gr < (unsigned)IMG_H && (unsigned)gc < (unsigned)IMG_W) {
      unsigned off = (unsigned)(gr * IMG_W + gc) * 4u;
      async_ld_f32(xb, off, sx_base + e);
      async_ld_f32(yb, off, sy_base + e);
    }
  }
  wait_async0();
  // Zero-fill halo lying outside the image (SAME zero padding), after async wait.
  for (int i = tid; i < LROWS * LCOLS; i += 256) {
    int row = i / LCOLS;
    int col = i - row * LCOLS;
    int gr = gy0 - 1 + row;
    int gc = gx0 - 1 + col;
    if (!((unsigned)gr < (unsigned)IMG_H && (unsigned)gc < (unsigned)IMG_W)) {
      sx[row * LSTR + col] = 0.0f;
      sy[row * LSTR + col] = 0.0f;
    }
  }
  __syncthreads();

  // 64 threads across tile width, 4 thread-rows; each walks 8 output rows with a
  // rolling 3-row register window of horizontally filtered channels.
  const int c  = tid & 63;        // tile column; LDS cols c..c+2 are (c-1..c+1)+halo
  const int r0 = (tid >> 6) * 8;  // first output tile row
  float hx[3], hy[3], hxx[3], hyy[3], hxy[3];
  hrow(sx, sy, r0,     c, hx[0], hy[0], hxx[0], hyy[0], hxy[0]); // tile row r0-1
  hrow(sx, sy, r0 + 1, c, hx[1], hy[1], hxx[1], hyy[1], hxy[1]); // tile row r0

  float acc = 0.0f;
#pragma unroll
  for (int k = 0; k < 8; ++k) {
    const int i0 = k % 3, i1 = (k + 1) % 3, i2 = (k + 2) % 3;
    hrow(sx, sy, r0 + k + 2, c, hx[i2], hy[i2], hxx[i2], hyy[i2], hxy[i2]);
    float mx  = GA * (hx[i0]  + hx[i2])  + GB * hx[i1];
    float my  = GA * (hy[i0]  + hy[i2])  + GB * hy[i1];
    float cxx = GA * (hxx[i0] + hxx[i2]) + GB * hxx[i1];
    float cyy = GA * (hyy[i0] + hyy[i2]) + GB * hyy[i1];
    float cxy = GA * (hxy[i0] + hxy[i2]) + GB * hxy[i1];
    float t = mx * my, qx = mx * mx, qy = my * my;
    float sxx = cxx - qx, syy = cyy - qy, sxy = cxy - t;
    float num = (2.0f * t + C1) * (2.0f * sxy + C2);
    float den = (qx + qy + C1) * (sxx + syy + C2);
    acc += num / den;
  }

  // Deterministic block reduction -> one partial per block.
  float s = acc;
  for (int o = 16; o; o >>= 1) s += __shfl_xor(s, o);
  if ((tid & 31) == 0) red[tid >> 5] = s;
  __syncthreads();
  if (tid == 0) {
    float tsum = 0.0f;
    for (int i = 0; i < 8; ++i) tsum += red[i];
    partial[blockIdx.x + NBX * (blockIdx.y + NBY * blockIdx.z)] = tsum;
  }
}

__global__ __launch_bounds__(256)
void ssim_reduce_kernel(const float* __restrict__ partial, float* __restrict__ out) {
  __shared__ float red[8];
  float acc = 0.0f;
  for (int i = threadIdx.x; i < NPART; i += 256) acc += partial[i];

  float wtot;
#if defined(__AMDGCN__) && __has_builtin(__builtin_amdgcn_wmma_f32_16x16x4_f32)
  // Cross-lane sum via V_WMMA_F32_16X16X4_F32: A holds the 32 per-lane accs
  // (VGPR0 slot; K=0 for lanes 0-15, K=2 for lanes 16-31), B = all-ones, so
  // D[m][n] = rowsum(A,m). Lane j holds D[0..7][j], lane j^16 holds D[8..15][j];
  // their sums add to the full wave total. f32 in/out -> no precision loss.
  v2f a; a.x = acc; a.y = 0.0f;
  v2f b; b.x = 1.0f; b.y = 1.0f;
  v8f cz = {};
  v8f d = __builtin_amdgcn_wmma_f32_16x16x4_f32(false, a, false, b, (short)0, cz,
                                                false, false);
  float s = d[0] + d[1] + d[2] + d[3] + d[4] + d[5] + d[6] + d[7];
  wtot = s + __shfl_xor(s, 16);
#else
  float s = acc;
  for (int o = 16; o; o >>= 1) s += __shfl_xor(s, o);
  wtot = s;
#endif
  if ((threadIdx.x & 31) == 0) red[threadIdx.x >> 5] = wtot;
  __syncthreads();
  if (threadIdx.x == 0) {
    float tsum = 0.0f;
    for (int i = 0; i < 8; ++i) tsum += red[i];
    out[0] = 1.0f - tsum * (1.0f / ((float)IMG_B * (float)IMG_H * (float)IMG_W));
  }
}

extern "C" void kernel_launch(void* const* d_in, const int* in_sizes, int n_in,
                              void* d_out, int out_size, void* d_ws, size_t ws_size,
                              hipStream_t stream) {
  (void)in_sizes; (void)n_in; (void)out_size; (void)ws_size;
  const float* x = (const float*)d_in[0];
  const float* y = (const float*)d_in[1];
  float* out     = (float*)d_out;
  float* partial = (float*)d_ws;  // NPART floats = 38.4 KB of scratch

  dim3 grid(NBX, NBY, IMG_B);     // 10 x 15 x 64 = 9600 blocks
  ssim_tile_kernel<<<grid, 256, 0, stream>>>(x, y, partial);
  ssim_reduce_kernel<<<1, 256, 0, stream>>>(partial, out);
}